// Gumbel_Sinkhorn_Network_39539468927344
// MI455X (gfx1250) — compile-verified
//
#include <hip/hip_runtime.h>
#include <hip/hip_bf16.h>
#include <math.h>

#define NN 4096
#define NITER 20

typedef __attribute__((ext_vector_type(2))) float v2f;
typedef __attribute__((ext_vector_type(8))) float v8f;
typedef int v4i __attribute__((vector_size(16)));  // matches builtin param type

#define AS3 __attribute__((address_space(3)))
#define AS1 __attribute__((address_space(1)))

// ---------------------------------------------------------------------------
// Stage the 4096-float v vector into LDS. Uses the CDNA5 async global->LDS
// path (ASYNCcnt) when the toolchain exposes it; falls back to plain copies.
// Must be called by all 256 threads of the block.
// ---------------------------------------------------------------------------
__device__ __forceinline__ void stage_v(const float* __restrict__ g, float* s) {
  const int t = threadIdx.x;  // blockDim.x == 256
#if __has_builtin(__builtin_amdgcn_global_load_async_to_lds_b128) && \
    __has_builtin(__builtin_amdgcn_s_wait_asynccnt)
  #pragma unroll
  for (int q = 0; q < 4; ++q) {
    const int e = q * 1024 + t * 4;  // contiguous 4 KB per instruction
    __builtin_amdgcn_global_load_async_to_lds_b128(
        (AS1 v4i*)(g + e), (AS3 v4i*)(s + e), 0, 0);
  }
  __builtin_amdgcn_s_wait_asynccnt(0);
#else
  const float4* g4 = (const float4*)g;
  float4*       s4 = (float4*)s;
  #pragma unroll
  for (int q = 0; q < 4; ++q) {
    const int e = q * 256 + t;
    s4[e] = g4[e];
  }
#endif
  __syncthreads();
}

// ---------------------------------------------------------------------------
// K = exp(W)   (16.7M elements, float4 per thread)
// ---------------------------------------------------------------------------
__global__ __launch_bounds__(256) void k_exp(const float* __restrict__ w,
                                             float* __restrict__ k) {
  const size_t i = (size_t)blockIdx.x * blockDim.x + threadIdx.x;
  const float4 a = ((const float4*)w)[i];
  float4 r;
  r.x = __expf(a.x);
  r.y = __expf(a.y);
  r.z = __expf(a.z);
  r.w = __expf(a.w);
  ((float4*)k)[i] = r;
}

__global__ __launch_bounds__(256) void k_init_u(float* __restrict__ u) {
  u[blockIdx.x * blockDim.x + threadIdx.x] = 1.0f;
}

// ---------------------------------------------------------------------------
// Column matvec partials: part[chunk][j] = sum_{i in chunk} u[i] * K[i][j].
// Each wave owns 16 columns and walks 512 rows in 4-row WMMA steps:
//   B[k][n] = u[i+k]*K[i+k][j0+n],  A = ones  =>  D[:, n] = partial colsum.
// Folding u into B and using A=1 makes the result invariant to the exact
// B slot->K mapping (sum over all four K slots).
// grid = (32, 8), block = 256 (8 waves); 32 blocks * 8 waves * 16 = 4096 cols, 8*512 rows.
// ---------------------------------------------------------------------------
__global__ __launch_bounds__(256) void k_colsum(const float* __restrict__ K,
                                                const float* __restrict__ u,
                                                float* __restrict__ part) {
  const int wave = threadIdx.x >> 5;
  const int lane = threadIdx.x & 31;
  const int half = lane >> 4;   // 0: K-slots {0,1}, 1: K-slots {2,3}
  const int l16  = lane & 15;
  const int col  = blockIdx.x * 128 + wave * 16 + l16;
  const int i0   = blockIdx.y * 512;

#if __has_builtin(__builtin_amdgcn_wmma_f32_16x16x4_f32)
  v8f acc = {0.f, 0.f, 0.f, 0.f, 0.f, 0.f, 0.f, 0.f};
  const v2f ones = {1.0f, 1.0f};
  for (int i = i0; i < i0 + 512; i += 4) {
    const int r0 = i + 2 * half;
    v2f b;
    b[0] = u[r0]     * K[(size_t)r0 * NN + col];
    b[1] = u[r0 + 1] * K[(size_t)(r0 + 1) * NN + col];
    acc = __builtin_amdgcn_wmma_f32_16x16x4_f32(
        false, ones, false, b, (short)0, acc, false, false);
  }
  // C/D layout: VGPR0, lanes 0-15 hold D[0][n] (all rows of D are equal).
  if (lane < 16) part[blockIdx.y * NN + col] = acc[0];
#else
  if (lane < 16) {
    float acc = 0.0f;
    for (int i = i0; i < i0 + 512; ++i)
      acc = fmaf(u[i], K[(size_t)i * NN + col], acc);
    part[blockIdx.y * NN + col] = acc;
  }
#endif
}

// v[j] = 1 / sum_c part[c][j]
__global__ __launch_bounds__(256) void k_colreduce(const float* __restrict__ part,
                                                   float* __restrict__ v) {
  const int j = blockIdx.x * blockDim.x + threadIdx.x;
  float s = 0.0f;
  #pragma unroll
  for (int c = 0; c < 8; ++c) s += part[c * NN + j];
  v[j] = 1.0f / s;
}

// ---------------------------------------------------------------------------
// Row matvec: u[r] = 1 / sum_j K[r][j] * v[j].  Wave per row, v in LDS.
// grid = 512, block = 256 (8 waves = 8 rows).
// ---------------------------------------------------------------------------
__global__ __launch_bounds__(256) void k_rowsum(const float* __restrict__ K,
                                                const float* __restrict__ v,
                                                float* __restrict__ u) {
  __shared__ float s_v[NN];
  stage_v(v, s_v);

  const int wave = threadIdx.x >> 5;
  const int lane = threadIdx.x & 31;
  const int r    = blockIdx.x * 8 + wave;

  const float4* k4 = (const float4*)(K + (size_t)r * NN);
  const float4* v4 = (const float4*)s_v;
  float acc = 0.0f;
  for (int t = lane; t < NN / 4; t += 32) {
    const float4 kk = k4[t];
    const float4 vv = v4[t];
    acc = fmaf(kk.x, vv.x, acc);
    acc = fmaf(kk.y, vv.y, acc);
    acc = fmaf(kk.z, vv.z, acc);
    acc = fmaf(kk.w, vv.w, acc);
  }
  #pragma unroll
  for (int off = 16; off > 0; off >>= 1) acc += __shfl_xor(acc, off, 32);
  if (lane == 0) u[r] = 1.0f / acc;
}

// ---------------------------------------------------------------------------
// Finalize: P = u_i * K_ij * v_j written in place over K; per-row argmax with
// first-max tiebreak; one-hot written to `hard`.  Wave per row.
// ---------------------------------------------------------------------------
__global__ __launch_bounds__(256) void k_finalize(float* __restrict__ KP,
                                                  const float* __restrict__ u,
                                                  const float* __restrict__ v,
                                                  float* __restrict__ hard) {
  __shared__ float s_v[NN];
  stage_v(v, s_v);

  const int wave = threadIdx.x >> 5;
  const int lane = threadIdx.x & 31;
  const int r    = blockIdx.x * 8 + wave;
  const float ui = u[r];

  float4*       k4 = (float4*)(KP + (size_t)r * NN);
  const float4* v4 = (const float4*)s_v;

  float m  = -INFINITY;
  int   mi = 0x7fffffff;
  for (int t = lane; t < NN / 4; t += 32) {
    const float4 kk = k4[t];
    const float4 vv = v4[t];
    float4 p;
    p.x = ui * kk.x * vv.x;
    p.y = ui * kk.y * vv.y;
    p.z = ui * kk.z * vv.z;
    p.w = ui * kk.w * vv.w;
    k4[t] = p;  // P_hat in place over K
    const int j = t * 4;
    if (p.x > m) { m = p.x; mi = j; }
    if (p.y > m) { m = p.y; mi = j + 1; }
    if (p.z > m) { m = p.z; mi = j + 2; }
    if (p.w > m) { m = p.w; mi = j + 3; }
  }
  // (max, lowest-index) butterfly combine -> all lanes hold the row argmax
  #pragma unroll
  for (int off = 16; off > 0; off >>= 1) {
    const float om = __shfl_xor(m, off, 32);
    const int   oi = __shfl_xor(mi, off, 32);
    if (om > m || (om == m && oi < mi)) { m = om; mi = oi; }
  }

  float4* h4 = (float4*)(hard + (size_t)r * NN);
  for (int t = lane; t < NN / 4; t += 32) {
    const int j = t * 4;
    float4 h;
    h.x = (j     == mi) ? 1.0f : 0.0f;
    h.y = (j + 1 == mi) ? 1.0f : 0.0f;
    h.z = (j + 2 == mi) ? 1.0f : 0.0f;
    h.w = (j + 3 == mi) ? 1.0f : 0.0f;
    h4[t] = h;
  }
}

// ---------------------------------------------------------------------------
extern "C" void kernel_launch(void* const* d_in, const int* in_sizes, int n_in,
                              void* d_out, int out_size, void* d_ws, size_t ws_size,
                              hipStream_t stream) {
  const float* W = (const float*)d_in[0];

  float* out  = (float*)d_out;
  float* K    = out;                       // N*N: K, then P_hat in place
  float* hard = out + (size_t)NN * NN;     // N*N: one-hot output

  float* u    = (float*)d_ws;              // 4096
  float* v    = u + NN;                    // 4096
  float* part = v + NN;                    // 8 * 4096 column partials

  // K = exp(W); u = 1
  k_exp<<<(NN * NN / 4) / 256, 256, 0, stream>>>(W, K);
  k_init_u<<<NN / 256, 256, 0, stream>>>(u);

  for (int it = 0; it < NITER; ++it) {
    // column normalize: v = 1 / (K^T u)
    k_colsum<<<dim3(32, 8), 256, 0, stream>>>(K, u, part);
    k_colreduce<<<NN / 256, 256, 0, stream>>>(part, v);
    // row normalize: u = 1 / (K v)
    k_rowsum<<<NN / 8, 256, 0, stream>>>(K, v, u);
  }

  // P_hat = diag(u) K diag(v) (in place), one-hot argmax into `hard`
  k_finalize<<<NN / 8, 256, 0, stream>>>(K, u, v, hard);
}